// Decoder_6691559047677
// MI455X (gfx1250) — compile-verified
//
#include <hip/hip_runtime.h>
#include <math.h>

// ---------------------------------------------------------------------------
// Greedy GRU decoder for MI455X (gfx1250, wave32).
//   B=32, H=E=16, V=32000, T=100 steps, output (B,T,V) f32 = 409.6 MB.
// Memory-bound: ~410 MB NT stores -> ~18us floor @ 23.3 TB/s. W_out (2MB)
// stays L2-resident. Serial feedback (argmax -> next token) => 2 kernels/step.
// Logits GEMM uses V_WMMA_F32_16X16X4_F32 chained over K=16.
// ---------------------------------------------------------------------------

typedef __attribute__((ext_vector_type(2))) float v2f;
typedef __attribute__((ext_vector_type(8))) float v8f;

#define B_   32
#define H_   16
#define E_   16
#define V_   32000
#define T_   100
#define NT_  (V_ / 16)      // 2000 column tiles of 16
#define WPB  8              // waves per block
#define NB_  (NT_ / WPB)    // 250 blocks

// Map float -> monotonically ordered u32 (for packed argmax keys).
__device__ __forceinline__ unsigned int ford(float f) {
    unsigned int u = __float_as_uint(f);
    return (u & 0x80000000u) ? ~u : (u | 0x80000000u);
}

__device__ __forceinline__ unsigned long long shflx64(unsigned long long x, int m) {
    int lo = __shfl_xor((int)(unsigned int)x, m, 32);
    int hi = __shfl_xor((int)(unsigned int)(x >> 32), m, 32);
    return ((unsigned long long)(unsigned int)hi << 32) | (unsigned int)lo;
}

__device__ __forceinline__ unsigned long long umax64(unsigned long long a, unsigned long long b) {
    return a > b ? a : b;
}

// ---------------------------------------------------------------------------
// Kernel A: one wave32. Lane b = batch row.
//  step==0 : tok=SOS, h = encoder_hidden[0][b]
//  step>0  : unpack argmax key from previous logits kernel, h from workspace
//  Then reset argmax buffer and run the PyTorch GRU cell; write h_new to ws.
// ---------------------------------------------------------------------------
__global__ __launch_bounds__(32) void gru_step_kernel(
    const float* __restrict__ enc_hidden,   // (1,B,H)
    const float* __restrict__ embedding,    // (V,E)
    const float* __restrict__ W_ih,         // (3H,E)
    const float* __restrict__ W_hh,         // (3H,H)
    const float* __restrict__ b_ih,         // (3H)
    const float* __restrict__ b_hh,         // (3H)
    float* __restrict__ h_ws,               // (B,H)
    unsigned long long* __restrict__ amax_ws, // (B)
    int step)
{
    const int b = threadIdx.x;
    if (b >= B_) return;

    int tok;
    float h[H_];
    if (step == 0) {
        tok = 0;  // SOS_TOKEN
        #pragma unroll
        for (int k = 0; k < H_; ++k) h[k] = enc_hidden[b * H_ + k];
    } else {
        unsigned long long key = amax_ws[b];
        tok = (int)(0x7FFFFFFFu - (unsigned int)(key & 0xFFFFFFFFu));
        #pragma unroll
        for (int k = 0; k < H_; ++k) h[k] = h_ws[b * H_ + k];
    }
    amax_ws[b] = 0ull;  // reset for this step's argmax (0 < any valid key)

    float x[E_];
    #pragma unroll
    for (int k = 0; k < E_; ++k) x[k] = embedding[(size_t)tok * E_ + k];

    float rg[H_], zg[H_];
    #pragma unroll
    for (int j = 0; j < H_; ++j) {           // reset gate (rows 0..15)
        float gi = b_ih[j], gh = b_hh[j];
        #pragma unroll
        for (int k = 0; k < E_; ++k) gi += x[k] * W_ih[j * E_ + k];
        #pragma unroll
        for (int k = 0; k < H_; ++k) gh += h[k] * W_hh[j * H_ + k];
        rg[j] = 1.0f / (1.0f + __expf(-(gi + gh)));
    }
    #pragma unroll
    for (int j = 0; j < H_; ++j) {           // update gate (rows 16..31)
        float gi = b_ih[H_ + j], gh = b_hh[H_ + j];
        #pragma unroll
        for (int k = 0; k < E_; ++k) gi += x[k] * W_ih[(H_ + j) * E_ + k];
        #pragma unroll
        for (int k = 0; k < H_; ++k) gh += h[k] * W_hh[(H_ + j) * H_ + k];
        zg[j] = 1.0f / (1.0f + __expf(-(gi + gh)));
    }
    #pragma unroll
    for (int j = 0; j < H_; ++j) {           // new gate (rows 32..47) + blend
        float gi = b_ih[2 * H_ + j], gh = b_hh[2 * H_ + j];
        #pragma unroll
        for (int k = 0; k < E_; ++k) gi += x[k] * W_ih[(2 * H_ + j) * E_ + k];
        #pragma unroll
        for (int k = 0; k < H_; ++k) gh += h[k] * W_hh[(2 * H_ + j) * H_ + k];
        float n = tanhf(gi + rg[j] * gh);
        h_ws[b * H_ + j] = (1.0f - zg[j]) * n + zg[j] * h[j];
    }
}

// ---------------------------------------------------------------------------
// Kernel B: logits = h_new @ W_out^T + b_out  via V_WMMA_F32_16X16X4_F32.
// Each wave owns a 32x16 output tile (two 16x16 WMMA tiles sharing B operand),
// K=16 -> 4 chained WMMAs per tile. NT stores to d_out; wave-shuffle + LDS +
// u64 atomicMax argmax reduction into amax_ws.
// ---------------------------------------------------------------------------
__global__ __launch_bounds__(256) void logits_kernel(
    const float* __restrict__ W_out,        // (V,H)
    const float* __restrict__ b_out,        // (V)
    const float* __restrict__ h_ws,         // (B,H)
    float* __restrict__ out,                // (B,T,V)
    unsigned long long* __restrict__ amax_ws,
    int step)
{
    __shared__ unsigned long long red[B_][WPB];

    const int wave  = threadIdx.x >> 5;
    const int lane  = threadIdx.x & 31;
    const int tile  = blockIdx.x * WPB + wave;       // 0..1999
    const int vbase = tile * 16;
    const int n     = lane & 15;                     // column within tile / A row
    const int khalf = lane >> 4;                     // K-half per ISA f32 layout

    // Prefetch next block's W_out tile row into cache (global_prefetch path).
    if (tile + WPB < NT_)
        __builtin_prefetch(W_out + (size_t)(vbase + WPB * 16 + n) * H_, 0, 0);

    // A (16x4 f32): lane m=n, VGPR0/1 = K = 4*kk + 2*khalf + {0,1}
    // B (4x16 f32): lane col n,  VGPR0/1 = K = 4*kk + 2*khalf + {0,1}
    v2f a0[4], a1[4], bf[4];
    #pragma unroll
    for (int kk = 0; kk < 4; ++kk) {
        const int ko = kk * 4 + khalf * 2;
        a0[kk] = *(const v2f*)(h_ws + n * H_ + ko);
        a1[kk] = *(const v2f*)(h_ws + (n + 16) * H_ + ko);
        bf[kk] = *(const v2f*)(W_out + (size_t)(vbase + n) * H_ + ko);
    }

    const float bias = b_out[vbase + n];             // same column for all 8 C VGPRs
    v8f c0, c1;
    #pragma unroll
    for (int i = 0; i < 8; ++i) { c0[i] = bias; c1[i] = bias; }

    #pragma unroll
    for (int kk = 0; kk < 4; ++kk)
        c0 = __builtin_amdgcn_wmma_f32_16x16x4_f32(false, a0[kk], false, bf[kk],
                                                   (short)0, c0, false, false);
    #pragma unroll
    for (int kk = 0; kk < 4; ++kk)
        c1 = __builtin_amdgcn_wmma_f32_16x16x4_f32(false, a1[kk], false, bf[kk],
                                                   (short)0, c1, false, false);

    const int v = vbase + n;
    const unsigned int vinv = 0x7FFFFFFFu - (unsigned int)v;  // tie -> smallest v wins

    #pragma unroll
    for (int r = 0; r < 8; ++r) {
        const int m0 = khalf ? (r + 8) : r;          // C/D layout: M=r / M=8+r per half
        const int m1 = m0 + 16;
        // Non-temporal stores: 409.6 MB stream must not evict L2-resident W_out.
        __builtin_nontemporal_store(c0[r], out + ((size_t)m0 * T_ + step) * V_ + v);
        __builtin_nontemporal_store(c1[r], out + ((size_t)m1 * T_ + step) * V_ + v);

        unsigned long long k0 = ((unsigned long long)ford(c0[r]) << 32) | vinv;
        unsigned long long k1 = ((unsigned long long)ford(c1[r]) << 32) | vinv;
        #pragma unroll
        for (int off = 1; off < 16; off <<= 1) {     // reduce within each 16-lane half
            k0 = umax64(k0, shflx64(k0, off));
            k1 = umax64(k1, shflx64(k1, off));
        }
        if (lane == 0)  { red[r][wave]      = k0; red[16 + r][wave] = k1; }
        if (lane == 16) { red[8 + r][wave]  = k0; red[24 + r][wave] = k1; }
    }

    __syncthreads();
    if (threadIdx.x < B_) {                          // one atomic per row per block
        unsigned long long best = red[threadIdx.x][0];
        #pragma unroll
        for (int w = 1; w < WPB; ++w) best = umax64(best, red[threadIdx.x][w]);
        atomicMax(&amax_ws[threadIdx.x], best);
    }
}

// ---------------------------------------------------------------------------
extern "C" void kernel_launch(void* const* d_in, const int* in_sizes, int n_in,
                              void* d_out, int out_size, void* d_ws, size_t ws_size,
                              hipStream_t stream) {
    // setup_inputs order (encoder_outputs unused by the reference decode path):
    const float* enc_hidden = (const float*)d_in[1];   // (1,B,H)
    const float* embedding  = (const float*)d_in[2];   // (V,E)
    const float* W_ih       = (const float*)d_in[3];   // (3H,E)
    const float* W_hh       = (const float*)d_in[4];   // (3H,H)
    const float* b_ih       = (const float*)d_in[5];   // (3H)
    const float* b_hh       = (const float*)d_in[6];   // (3H)
    const float* W_out      = (const float*)d_in[7];   // (V,H)
    const float* b_out      = (const float*)d_in[8];   // (V)
    float* out = (float*)d_out;

    float* h_ws = (float*)d_ws;                                     // B*H f32 = 2 KB
    unsigned long long* amax_ws =
        (unsigned long long*)((char*)d_ws + B_ * H_ * sizeof(float)); // B u64 = 256 B

    for (int t = 0; t < T_; ++t) {
        gru_step_kernel<<<1, 32, 0, stream>>>(enc_hidden, embedding, W_ih, W_hh,
                                              b_ih, b_hh, h_ws, amax_ws, t);
        logits_kernel<<<NB_, 256, 0, stream>>>(W_out, b_out, h_ws, out, amax_ws, t);
    }
}